// CartesianProductClassifier_77867757077200
// MI455X (gfx1250) — compile-verified
//
#include <hip/hip_runtime.h>
#include <hip/hip_bf16.h>
#include <math.h>

typedef __attribute__((ext_vector_type(2))) float v2f;
typedef __attribute__((ext_vector_type(4))) float v4f;
typedef __attribute__((ext_vector_type(8))) float v8f;

#define BATCH 8
#define N_PTS 512
#define DIN   32
#define H1DIM 64
#define H2DIM 64
#define H3DIM 32

// Compiler/scheduling fence so wave-private LDS producer->consumer (cross-lane,
// same wave) is not reordered. Hardware keeps DS ops in order per wave.
__device__ __forceinline__ void wave_sync() {
    __builtin_amdgcn_fence(__ATOMIC_ACQ_REL, "workgroup");
    __builtin_amdgcn_wave_barrier();
}

// ---------------------------------------------------------------------------
// Kernel 1: factorized layer 1.
//   Apre[row, c] = sum_d x[row, d] * W1[d, c]            (row = b*N + j)
//   Bpre[row, c] = sum_d x[row, d] * W1[32 + d, c] + b1  (row = b*N + i)
// ---------------------------------------------------------------------------
__global__ __launch_bounds__(64) void cpc_precompute_kernel(
        const float* __restrict__ x, const float* __restrict__ W1,
        const float* __restrict__ b1,
        float* __restrict__ Apre, float* __restrict__ Bpre) {
    __shared__ float xr[DIN];
    const int row = blockIdx.x;      // b*N + n
    const int c = threadIdx.x;       // 0..63
    if (c < DIN) xr[c] = x[row * DIN + c];
    __syncthreads();
    float a = 0.f;
    float bacc = b1[c];
    #pragma unroll
    for (int d = 0; d < DIN; ++d) {
        const float xv = xr[d];
        a    = fmaf(xv, W1[d * H1DIM + c], a);
        bacc = fmaf(xv, W1[(DIN + d) * H1DIM + c], bacc);
    }
    Apre[row * H1DIM + c] = a;
    Bpre[row * H1DIM + c] = bacc;
}

// ---------------------------------------------------------------------------
// Kernel 2: one workgroup per (b, i). 8 waves; each wave owns 16-row j-tiles.
// Layers 2/3 on V_WMMA_F32_16X16X4_F32 (full fp32 matrix path).
// ---------------------------------------------------------------------------
#define KSTR 68   // padded K-stride (floats): bank = (4*row + k) % 64, conflict-free
#define TSTR 36   // padded stride for the 32-wide H3 staging tile

__global__ __launch_bounds__(256) void cpc_pair_mlp_kernel(
        const float* __restrict__ Apre, const float* __restrict__ Bpre,
        const float* __restrict__ W2, const float* __restrict__ b2,
        const float* __restrict__ W3, const float* __restrict__ b3,
        const float* __restrict__ W4, const float* __restrict__ b4,
        float* __restrict__ out) {
    __shared__ float W2Ts[H2DIM * KSTR];   // [n][k] = W2[k][n], n<64, k<64
    __shared__ float W3Ts[H3DIM * KSTR];   // [n][k] = W3[k][n], n<32, k<64
    __shared__ float b2s[H2DIM];
    __shared__ float b3s[H3DIM];
    __shared__ float W4s[H3DIM];
    __shared__ float Bis[H1DIM];           // Bpre row for this (b,i), +b1 folded
    __shared__ float bufA[8][16 * KSTR];   // per-wave: H1 staging, then H2 (aliased)
    __shared__ float bufB[8][16 * TSTR];   // per-wave: H3 staging

    const int tid  = threadIdx.x;
    const int wave = tid >> 5;
    const int lane = tid & 31;
    const int half = lane >> 4;            // 0: lanes 0-15, 1: lanes 16-31
    const int mrow = lane & 15;
    const int bi   = blockIdx.x;           // b*N + i
    const int b    = bi >> 9;

    // ---- cooperative staging of shared operands ----
    for (int idx = tid; idx < H1DIM * H2DIM; idx += 256) {
        const int k = idx >> 6, n = idx & 63;          // W2[k][n]
        W2Ts[n * KSTR + k] = W2[idx];
    }
    for (int idx = tid; idx < H2DIM * H3DIM; idx += 256) {
        const int k = idx >> 5, n = idx & 31;          // W3[k][n]
        W3Ts[n * KSTR + k] = W3[idx];
    }
    if (tid < H2DIM) b2s[tid] = b2[tid];
    if (tid < H3DIM) { b3s[tid] = b3[tid]; W4s[tid] = W4[tid]; }
    if (tid < H1DIM) Bis[tid] = Bpre[bi * H1DIM + tid];
    __syncthreads();

    const float b4v = b4[0];
    float* H12 = &bufA[wave][0];
    float* H3s = &bufB[wave][0];

    for (int tt = 0; tt < 4; ++tt) {
        const int j0 = (wave + tt * 8) * 16;
        const float* arow = &Apre[(b * N_PTS + j0) * H1DIM];   // 16x64 contiguous

        // ---- layer 1: H1 = relu(Apre[j] + Bi) into wave-private LDS ----
        #pragma unroll
        for (int t = 0; t < 8; ++t) {
            const int li = lane + 32 * t;      // 0..255 float4 slots
            const int r  = li >> 4;            // tile row 0..15
            const int c  = 4 * (lane & 15);    // col 0..60 step 4
            v4f v = *(const v4f*)&arow[li * 4];
            const v4f bb = *(const v4f*)&Bis[c];
            v = v + bb;
            v.x = fmaxf(v.x, 0.f); v.y = fmaxf(v.y, 0.f);
            v.z = fmaxf(v.z, 0.f); v.w = fmaxf(v.w, 0.f);
            *(v4f*)&H12[r * KSTR + c] = v;
        }
        wave_sync();

        // ---- layer 2: H2 = relu(H1 @ W2 + b2), 16x64 tile via fp32 WMMA ----
        v8f acc[4] = {};
        #pragma unroll
        for (int k0 = 0; k0 < H1DIM; k0 += 4) {
            const v2f a = *(const v2f*)&H12[mrow * KSTR + k0 + 2 * half];
            #pragma unroll
            for (int nt = 0; nt < 4; ++nt) {
                const v2f bf =
                    *(const v2f*)&W2Ts[(nt * 16 + mrow) * KSTR + k0 + 2 * half];
                acc[nt] = __builtin_amdgcn_wmma_f32_16x16x4_f32(
                    false, a, false, bf, (short)0, acc[nt], false, false);
            }
        }
        wave_sync();
        #pragma unroll
        for (int r = 0; r < 8; ++r) {
            const int row = r + 8 * half;      // C/D layout: vgpr r, lane half
            #pragma unroll
            for (int nt = 0; nt < 4; ++nt) {
                const int col = nt * 16 + mrow;
                H12[row * KSTR + col] = fmaxf(acc[nt][r] + b2s[col], 0.f);
            }
        }
        wave_sync();

        // ---- layer 3: H3 = relu(H2 @ W3 + b3), 16x32 tile ----
        v8f c3[2] = {};
        #pragma unroll
        for (int k0 = 0; k0 < H2DIM; k0 += 4) {
            const v2f a = *(const v2f*)&H12[mrow * KSTR + k0 + 2 * half];
            #pragma unroll
            for (int nt = 0; nt < 2; ++nt) {
                const v2f bf =
                    *(const v2f*)&W3Ts[(nt * 16 + mrow) * KSTR + k0 + 2 * half];
                c3[nt] = __builtin_amdgcn_wmma_f32_16x16x4_f32(
                    false, a, false, bf, (short)0, c3[nt], false, false);
            }
        }
        wave_sync();
        #pragma unroll
        for (int r = 0; r < 8; ++r) {
            const int row = r + 8 * half;
            #pragma unroll
            for (int nt = 0; nt < 2; ++nt) {
                const int col = nt * 16 + mrow;
                H3s[row * TSTR + col] = fmaxf(c3[nt][r] + b3s[col], 0.f);
            }
        }
        wave_sync();

        // ---- layer 4: out = sigmoid(H3 @ W4 + b4); split-K across lane halves ----
        float s = 0.f;
        #pragma unroll
        for (int k = 0; k < 16; ++k) {
            const int kk = half * 16 + k;
            s = fmaf(H3s[mrow * TSTR + kk], W4s[kk], s);
        }
        s += __shfl_xor(s, 16, 32);
        if (half == 0) {
            const float z = s + b4v;
            out[bi * N_PTS + j0 + mrow] = 1.f / (1.f + __expf(-z));
        }
        wave_sync();   // next iteration overwrites H12/H3s
    }
}

// ---------------------------------------------------------------------------
extern "C" void kernel_launch(void* const* d_in, const int* in_sizes, int n_in,
                              void* d_out, int out_size, void* d_ws, size_t ws_size,
                              hipStream_t stream) {
    const float* x  = (const float*)d_in[0];
    const float* W1 = (const float*)d_in[1];
    const float* b1 = (const float*)d_in[2];
    const float* W2 = (const float*)d_in[3];
    const float* b2 = (const float*)d_in[4];
    const float* W3 = (const float*)d_in[5];
    const float* b3 = (const float*)d_in[6];
    const float* W4 = (const float*)d_in[7];
    const float* b4 = (const float*)d_in[8];
    float* out = (float*)d_out;

    float* Apre = (float*)d_ws;                      // [B*N, 64]  (1 MB)
    float* Bpre = Apre + BATCH * N_PTS * H1DIM;      // [B*N, 64]  (1 MB)

    cpc_precompute_kernel<<<BATCH * N_PTS, 64, 0, stream>>>(x, W1, b1, Apre, Bpre);
    cpc_pair_mlp_kernel<<<BATCH * N_PTS, 256, 0, stream>>>(
        Apre, Bpre, W2, b2, W3, b3, W4, b4, out);
}